// center_align_68564857913608
// MI455X (gfx1250) — compile-verified
//
#include <hip/hip_runtime.h>
#include <math.h>

// Problem constants (from the reference).
#define B_    8
#define C_    256
#define H_    96
#define W_    320
#define A_    36
#define HW_   (H_ * W_)
#define NPIX_ (B_ * HW_)   // 245760
#define NTILE 32           // pixels per block tile (W_ % 32 == 0)

// LDS B-tile: two 16-pixel planes. Plane 1 base is shifted by +32 dwords so the
// two planes' even-bank footprints are disjoint (0..30 vs 32..62) -> conflict-free.
#define PLANE1_OFF 4128    // floats (4096 data + 32 pad)
#define BS_FLOATS  (PLANE1_OFF + 4096)

typedef __attribute__((ext_vector_type(2))) float v2f;
typedef __attribute__((ext_vector_type(8))) float v8f;

// ---------------------------------------------------------------------------
// Pass 1: per-pixel top-3 over anchors, softmax blend, offsets + mask.
// ---------------------------------------------------------------------------
__global__ __launch_bounds__(256) void ca_offsets_kernel(
    const float* __restrict__ bbox_x, const float* __restrict__ bbox_y,
    const float* __restrict__ prob,
    const float* __restrict__ anchors_w, const float* __restrict__ anchors_h,
    float* __restrict__ offx, float* __restrict__ offy, float* __restrict__ mask)
{
    const int p = blockIdx.x * 256 + threadIdx.x;
    if (p >= NPIX_) return;
    const int b  = p / HW_;
    const int hw = p - b * HW_;
    const float* pr = prob + (size_t)b * A_ * HW_ + hw;

    float v0 = -INFINITY, v1 = -INFINITY, v2 = -INFINITY;
    int   i0 = 0, i1 = 0, i2 = 0;
    #pragma unroll
    for (int a = 0; a < A_; ++a) {
        const float pv = pr[(size_t)a * HW_];
        if (pv > v0)      { v2 = v1; i2 = i1; v1 = v0; i1 = i0; v0 = pv; i0 = a; }
        else if (pv > v1) { v2 = v1; i2 = i1; v1 = pv; i1 = a; }
        else if (pv > v2) { v2 = pv; i2 = a; }
    }

    // softmax over the top-3 (numerically shifted by v0)
    const float e0 = 1.0f;
    const float e1 = expf(v1 - v0);
    const float e2 = expf(v2 - v0);
    const float inv = 1.0f / (e0 + e1 + e2);
    const float s0 = e0 * inv, s1 = e1 * inv, s2 = e2 * inv;

    const float hard = (v0 > 0.5f) ? 1.0f : 0.0f;

    const float* bx = bbox_x + (size_t)b * A_ * HW_ + hw;
    const float* by = bbox_y + (size_t)b * A_ * HW_ + hw;

    const float ox = hard * (s0 * bx[(size_t)i0 * HW_] * 0.1f * anchors_w[i0] +
                             s1 * bx[(size_t)i1 * HW_] * 0.1f * anchors_w[i1] +
                             s2 * bx[(size_t)i2 * HW_] * 0.1f * anchors_w[i2]);
    const float oy = hard * (s0 * by[(size_t)i0 * HW_] * 0.1f * anchors_h[i0] +
                             s1 * by[(size_t)i1 * HW_] * 0.1f * anchors_h[i1] +
                             s2 * by[(size_t)i2 * HW_] * 0.1f * anchors_h[i2]);

    offx[p] = ox;
    offy[p] = oy;
    mask[p] = v0;
}

// ---------------------------------------------------------------------------
// Pass 2: fused bilinear-sample + 1x1-conv GEMM (f32 WMMA) + bias + residual.
// Block = 256 threads = 8 waves; tile of NTILE=32 consecutive-w pixels.
// Each wave owns 32 output rows x 32 pixels: 2 M-subtiles x 2 N-subtiles,
// so every A fragment feeds two v_wmma (halves weight L2 traffic per FLOP).
// ---------------------------------------------------------------------------
__global__ __launch_bounds__(256) void ca_fused_gemm_kernel(
    const float* __restrict__ x, const float* __restrict__ weight,
    const float* __restrict__ bias,
    const float* __restrict__ offx, const float* __restrict__ offy,
    const float* __restrict__ mask,
    float* __restrict__ out)
{
    // Element (k, n): plane = n>>4, addr = plane*PLANE1_OFF + (k>>1)*32 + (n&15)*2 + (k&1)
    // -> a WMMA B fragment {k, k+1} for lane n is one aligned 8-byte LDS load.
    __shared__ float Bs[BS_FLOATS];

    const int tid = threadIdx.x;
    const int p0  = blockIdx.x * NTILE;
    const int b   = p0 / HW_;
    const int hw0 = p0 - b * HW_;
    const int h   = hw0 / W_;
    const int w0c = hw0 - h * W_;

    const int wv   = tid >> 5;
    const int lane = tid & 31;
    const int half = lane >> 4;   // K split for A/B fragments (lane halves)
    const int lm   = lane & 15;   // M index in A frag / N index in B frag
    const int m0   = wv << 5;     // wave owns output rows [m0, m0+32)

    const float* wrow0 = weight + (size_t)(m0 + lm) * C_;  // rows m0..m0+15
    const float* wrow1 = wrow0 + (size_t)16 * C_;          // rows m0+16..m0+31
    // Warm L2/L0 with this wave's weight rows (global_prefetch_b8) while the
    // sampled-tile build phase below covers the latency.
    __builtin_prefetch(wrow0, 0, 0);
    __builtin_prefetch(wrow1, 0, 0);

    // ---- build sampled tile (bilinear gather, validity folded into weights) ----
    {
        const int px = tid & 31;   // pixel within tile (consecutive w -> coalesced gathers)
        const int cg = tid >> 5;   // channel group 0..7, each covers 32 channels
        const int p  = p0 + px;

        const float ox  = offx[p];
        const float oy  = offy[p];
        const float mk  = mask[p];
        const float xs  = (float)(w0c + px) + ox;
        const float ysv = (float)h + oy;
        const float x0f = floorf(xs), y0f = floorf(ysv);
        const float lx = xs - x0f,   ly = ysv - y0f;
        const int xi = (int)x0f, yi = (int)y0f;

        float f00 = (1.0f - ly) * (1.0f - lx);
        float f01 = (1.0f - ly) * lx;
        float f10 = ly * (1.0f - lx);
        float f11 = ly * lx;
        // fold out-of-bounds validity into the weights (ref zeroes OOB taps)
        const bool vx0 = (xi >= 0) & (xi < W_);
        const bool vx1 = (xi + 1 >= 0) & (xi + 1 < W_);
        const bool vy0 = (yi >= 0) & (yi < H_);
        const bool vy1 = (yi + 1 >= 0) & (yi + 1 < H_);
        f00 = (vy0 & vx0) ? f00 : 0.0f;
        f01 = (vy0 & vx1) ? f01 : 0.0f;
        f10 = (vy1 & vx0) ? f10 : 0.0f;
        f11 = (vy1 & vx1) ? f11 : 0.0f;

        const int xc0 = min(max(xi, 0), W_ - 1);
        const int xc1 = min(max(xi + 1, 0), W_ - 1);
        const int yc0 = min(max(yi, 0), H_ - 1);
        const int yc1 = min(max(yi + 1, 0), H_ - 1);
        const int o00 = yc0 * W_ + xc0, o01 = yc0 * W_ + xc1;
        const int o10 = yc1 * W_ + xc0, o11 = yc1 * W_ + xc1;

        const size_t base = (size_t)b * C_ * HW_;
        const int wbase = (px >> 4) * PLANE1_OFF + ((px & 15) << 1);
        #pragma unroll
        for (int j = 0; j < 32; ++j) {
            const int c = (cg << 5) + j;
            const float* xb = x + base + (size_t)c * HW_;
            const float sv = f00 * xb[o00] + f01 * xb[o01] +
                             f10 * xb[o10] + f11 * xb[o11];
            Bs[wbase + ((c >> 1) << 5) + (c & 1)] = sv * mk;
        }
    }
    __syncthreads();

    // ---- WMMA GEMM: 64 K-steps, 4 v_wmma per step (2 M-subtiles x 2 N-subtiles) ----
    v8f acc00 = {};
    v8f acc01 = {};
    v8f acc10 = {};
    v8f acc11 = {};

    #pragma unroll 8
    for (int k = 0; k < C_; k += 4) {
        const int kk = k + (half << 1);                   // {k, k+2} per lane half
        const v2f a0 = *(const v2f*)(wrow0 + kk);         // weight[m][kk..kk+1]
        const v2f a1 = *(const v2f*)(wrow1 + kk);
        const int bo = ((kk >> 1) << 5) + (lm << 1);
        const v2f b0 = *(const v2f*)(&Bs[bo]);            // sampled[kk..kk+1][lm]
        const v2f b1 = *(const v2f*)(&Bs[PLANE1_OFF + bo]); // sampled[..][lm+16]
        acc00 = __builtin_amdgcn_wmma_f32_16x16x4_f32(false, a0, false, b0,
                                                      (short)0, acc00, false, false);
        acc01 = __builtin_amdgcn_wmma_f32_16x16x4_f32(false, a0, false, b1,
                                                      (short)0, acc01, false, false);
        acc10 = __builtin_amdgcn_wmma_f32_16x16x4_f32(false, a1, false, b0,
                                                      (short)0, acc10, false, false);
        acc11 = __builtin_amdgcn_wmma_f32_16x16x4_f32(false, a1, false, b1,
                                                      (short)0, acc11, false, false);
    }

    // ---- epilogue: D layout (VGPR v -> M = v or v+8 per lane half), + bias + x ----
    const int n0 = lm;          // N-subtile 0 column
    const int n1 = lm + 16;     // N-subtile 1 column
    const size_t obase0 = (size_t)b * C_ * HW_ + (size_t)h * W_ + (w0c + n0);
    const size_t obase1 = (size_t)b * C_ * HW_ + (size_t)h * W_ + (w0c + n1);
    #pragma unroll
    for (int v = 0; v < 8; ++v) {
        const int mg0 = m0 + v + (half << 3);
        const int mg1 = mg0 + 16;
        const float bb0 = bias[mg0];
        const float bb1 = bias[mg1];
        const size_t o00 = obase0 + (size_t)mg0 * HW_;
        const size_t o01 = obase1 + (size_t)mg0 * HW_;
        const size_t o10 = obase0 + (size_t)mg1 * HW_;
        const size_t o11 = obase1 + (size_t)mg1 * HW_;
        out[o00] = acc00[v] + bb0 + x[o00];
        out[o01] = acc01[v] + bb0 + x[o01];
        out[o10] = acc10[v] + bb1 + x[o10];
        out[o11] = acc11[v] + bb1 + x[o11];
    }
}

// ---------------------------------------------------------------------------
extern "C" void kernel_launch(void* const* d_in, const int* in_sizes, int n_in,
                              void* d_out, int out_size, void* d_ws, size_t ws_size,
                              hipStream_t stream) {
    const float* x       = (const float*)d_in[0];
    const float* bbox_x  = (const float*)d_in[1];
    const float* bbox_y  = (const float*)d_in[2];
    const float* prob    = (const float*)d_in[3];
    const float* anch_w  = (const float*)d_in[4];
    const float* anch_h  = (const float*)d_in[5];
    const float* weight  = (const float*)d_in[6];
    const float* bias    = (const float*)d_in[7];
    float* out = (float*)d_out;

    float* ws   = (float*)d_ws;
    float* offx = ws;
    float* offy = ws + NPIX_;
    float* mask = ws + 2 * (size_t)NPIX_;

    ca_offsets_kernel<<<(NPIX_ + 255) / 256, 256, 0, stream>>>(
        bbox_x, bbox_y, prob, anch_w, anch_h, offx, offy, mask);

    ca_fused_gemm_kernel<<<NPIX_ / NTILE, 256, 0, stream>>>(
        x, weight, bias, offx, offy, mask, out);
}